// GCNModule_58566174048909
// MI455X (gfx1250) — compile-verified
//
#include <hip/hip_runtime.h>
#include <math.h>

// Problem constants (from the reference)
#define BB 64
#define TT 100
#define NN 25
#define VV 128
#define CC 128
#define EE 64
#define GG (BB * TT)          // 6400 graphs
#define GPW 8                 // graphs per workgroup
#define NWG (GG / GPW)        // 800 workgroups
#define EPSV 1e-5f

#define LDS_WSTRIDE 136       // 2*136 mod 64 = 16 -> lane-halves hit disjoint banks
#define LDS_XSTRIDE 132       // (m*132+k) mod 64 = (4m+k) mod 64, distinct per row

typedef __attribute__((ext_vector_type(2))) float v2f;
typedef __attribute__((ext_vector_type(8))) float v8f;
typedef int async_v4i __attribute__((vector_size(4 * sizeof(int))));

#define AS1 __attribute__((address_space(1)))
#define AS3 __attribute__((address_space(3)))

#if __has_builtin(__builtin_amdgcn_global_load_async_to_lds_b128)
#define HAVE_ASYNC 1
#else
#define HAVE_ASYNC 0
#endif

#if __has_builtin(__builtin_amdgcn_s_wait_asynccnt)
#define WAIT_ASYNC(n) __builtin_amdgcn_s_wait_asynccnt(n)
#else
#define WAIT_ASYNC(n) asm volatile("s_wait_asynccnt " #n ::: "memory")
#endif

// ---------------------------------------------------------------------------
// Kernel 0: densify normalized adjacency  A[t][s] = dis[s]*dis[t] (+ self loops)
// ---------------------------------------------------------------------------
__global__ void build_adj(const int* __restrict__ ei, float* __restrict__ Ag) {
    if (threadIdx.x != 0 || blockIdx.x != 0) return;
    float deg[NN];
    for (int n = 0; n < NN; ++n) deg[n] = 1.0f;          // self loop
    for (int e = 0; e < EE; ++e) {
        int t = ei[EE + e];
        if (t >= 0 && t < NN) deg[t] += 1.0f;
    }
    float dis[NN];
    for (int n = 0; n < NN; ++n) dis[n] = rsqrtf(deg[n]);
    for (int i = 0; i < NN * NN; ++i) Ag[i] = 0.0f;
    for (int e = 0; e < EE; ++e) {
        int s = ei[e], t = ei[EE + e];
        if (s >= 0 && s < NN && t >= 0 && t < NN)
            Ag[t * NN + s] += dis[s] * dis[t];
    }
    for (int n = 0; n < NN; ++n) Ag[n * NN + n] += dis[n] * dis[n];
}

// ---------------------------------------------------------------------------
// 25x128 @ 128x128 GEMM tile via V_WMMA_F32_16X16X4_F32.
// Fragment layouts per CDNA5 ISA 7.12.2:
//   A (16x4):  lane l holds row m=l&15; VGPR j holds K = k + j + 2*(l>>4)
//   B (4x16):  lane l holds col n=l&15; VGPR j holds K = k + j + 2*(l>>4)
//   D (16x16): VGPR i -> row i + 8*(l>>4), col l&15
// ---------------------------------------------------------------------------
__device__ __forceinline__ void gemm_25x128x128(const float* Xs, const float* Ws,
                                                float* Ms, int n0, int l15, int half) {
    v8f acc0 = {};
    v8f acc1 = {};
    const float* xr0 = Xs + l15 * LDS_XSTRIDE;           // rows 0..15
    const float* xr1 = Xs + (16 + l15) * LDS_XSTRIDE;    // rows 16..31 (25..31 pre-zeroed)
    const float* wc  = Ws + n0 + l15;
    const int koff = 2 * half;
#pragma unroll 8
    for (int k = 0; k < VV; k += 4) {
        const int ka = k + koff;
        v2f a0; a0.x = xr0[ka]; a0.y = xr0[ka + 1];
        v2f a1; a1.x = xr1[ka]; a1.y = xr1[ka + 1];
        v2f bf; bf.x = wc[ka * LDS_WSTRIDE]; bf.y = wc[(ka + 1) * LDS_WSTRIDE];
        acc0 = __builtin_amdgcn_wmma_f32_16x16x4_f32(false, a0, false, bf,
                                                     (short)0, acc0, false, false);
        acc1 = __builtin_amdgcn_wmma_f32_16x16x4_f32(false, a1, false, bf,
                                                     (short)0, acc1, false, false);
    }
#pragma unroll
    for (int i = 0; i < 8; ++i) {
        const int r0 = i + 8 * half;                     // 0..15
        Ms[r0 * LDS_XSTRIDE + n0 + l15] = acc0[i];
        const int r1 = 16 + i + 8 * half;                // 16..31
        if (r1 < NN) Ms[r1 * LDS_XSTRIDE + n0 + l15] = acc1[i];
    }
}

// ---------------------------------------------------------------------------
// Kernel 1: fused 2-layer GCN per graph; raw y -> d_out, per-graph BN partials -> ws
// Double-buffered node features with async global->LDS copies (ASYNCcnt).
// ---------------------------------------------------------------------------
__global__ __launch_bounds__(256) void gcn_fused(
    const float* __restrict__ x,  const float* __restrict__ W1,
    const float* __restrict__ b1, const float* __restrict__ W2,
    const float* __restrict__ b2, const float* __restrict__ Ag,
    float* __restrict__ y, float* __restrict__ psum, float* __restrict__ psumsq) {

    __shared__ float W1s[VV * LDS_WSTRIDE];   // 128x136
    __shared__ float W2s[VV * LDS_WSTRIDE];
    __shared__ float Xs[2][32 * LDS_XSTRIDE]; // double buffer; rows 25..31 stay zero
    __shared__ float Ms[32 * LDS_XSTRIDE];
    __shared__ float As[NN * NN];
    __shared__ float b1s[CC], b2s[CC];
    __shared__ float red[512];

    const int tid  = threadIdx.x;
    const int lane = tid & 31;
    const int wv   = tid >> 5;       // wave 0..7 -> owns cols n0..n0+15
    const int half = lane >> 4;
    const int l15  = lane & 15;
    const int n0   = wv * 16;
    const int gbase = blockIdx.x * GPW;

    // Stage weights (float4, LDS row start 544B -> 16B aligned)
    for (int i = tid; i < VV * (CC / 4); i += 256) {
        const int r = i >> 5, c4 = i & 31;
        float4 v1 = ((const float4*)W1)[i];
        float4 v2 = ((const float4*)W2)[i];
        *(float4*)&W1s[r * LDS_WSTRIDE + c4 * 4] = v1;
        *(float4*)&W2s[r * LDS_WSTRIDE + c4 * 4] = v2;
    }
    for (int i = tid; i < NN * NN; i += 256) As[i] = Ag[i];
    if (tid < CC) { b1s[tid] = b1[tid]; b2s[tid] = b2[tid]; }
    // zero padding rows 25..31 of both X buffers (aggregation only writes rows < 25)
    for (int i = tid; i < 7 * LDS_XSTRIDE; i += 256) {
        Xs[0][NN * LDS_XSTRIDE + i] = 0.0f;
        Xs[1][NN * LDS_XSTRIDE + i] = 0.0f;
    }

    // issue node-feature tile copy for graph g into buffer b
    auto issue_x_load = [&](int g, int b) {
        const float* xg = x + (size_t)g * (NN * VV);
        for (int i = tid; i < NN * (VV / 4); i += 256) {   // 800 float4; wave0: 4, rest: 3
            const int r = i >> 5, c4 = i & 31;
#if HAVE_ASYNC
            __builtin_amdgcn_global_load_async_to_lds_b128(
                (AS1 async_v4i*)(unsigned long long)(const void*)(xg + (size_t)i * 4),
                (AS3 async_v4i*)&Xs[b][r * LDS_XSTRIDE + c4 * 4],
                0, 0);
#else
            float4 v = ((const float4*)xg)[i];
            *(float4*)&Xs[b][r * LDS_XSTRIDE + c4 * 4] = v;
#endif
        }
    };

    issue_x_load(gbase, 0);                      // prologue: prefetch graph 0

    int buf = 0;
    for (int gi = 0; gi < GPW; ++gi) {
        const int g = gbase + gi;

        // overlap: start next graph's copy into the other buffer
        if (gi + 1 < GPW) issue_x_load(g + 1, buf ^ 1);
#if HAVE_ASYNC
        // async loads complete in order -> waiting down to "next's count"
        // guarantees the current buffer's copies have landed
        if (gi + 1 < GPW) {
            if (wv == 0) { WAIT_ASYNC(4); } else { WAIT_ASYNC(3); }
        } else {
            WAIT_ASYNC(0);
        }
#endif
        __syncthreads();                          // publish Xs[buf]

        // ---- layer 1: M = X @ W1 ; X <- A@M + b1 ----
        gemm_25x128x128(Xs[buf], W1s, Ms, n0, l15, half);
        __syncthreads();
        for (int idx = tid; idx < NN * CC; idx += 256) {
            const int t = idx >> 7, c = idx & 127;
            float acc = b1s[c];
#pragma unroll
            for (int s = 0; s < NN; ++s)
                acc = fmaf(As[t * NN + s], Ms[s * LDS_XSTRIDE + c], acc);
            Xs[buf][t * LDS_XSTRIDE + c] = acc;
        }
        __syncthreads();

        // ---- layer 2: M = H1 @ W2 ; y = A@M + b2 (raw, pre-BN) ----
        gemm_25x128x128(Xs[buf], W2s, Ms, n0, l15, half);
        __syncthreads();
        float ls = 0.0f, lss = 0.0f;
        for (int idx = tid; idx < NN * CC; idx += 256) {
            const int t = idx >> 7, c = idx & 127;
            float acc = b2s[c];
#pragma unroll
            for (int s = 0; s < NN; ++s)
                acc = fmaf(As[t * NN + s], Ms[s * LDS_XSTRIDE + c], acc);
            y[(size_t)g * (NN * CC) + idx] = acc;
            ls += acc;
            lss = fmaf(acc, acc, lss);
        }
        // deterministic block reduction of BN partials
        red[tid] = ls; red[256 + tid] = lss;
        __syncthreads();
        for (int off = 128; off > 0; off >>= 1) {
            if (tid < off) {
                red[tid]       += red[tid + off];
                red[256 + tid] += red[256 + tid + off];
            }
            __syncthreads();
        }
        if (tid == 0) { psum[g] = red[0]; psumsq[g] = red[256]; }
        __syncthreads();

        buf ^= 1;
    }
}

// ---------------------------------------------------------------------------
// Kernel 2: per-T mean / inv-std from the 64 per-graph partials
// ---------------------------------------------------------------------------
__global__ void bn_stats(const float* __restrict__ psum, const float* __restrict__ psumsq,
                         float* __restrict__ meanv, float* __restrict__ invv) {
    const int t = blockIdx.x * blockDim.x + threadIdx.x;
    if (t >= TT) return;
    float S = 0.0f, SS = 0.0f;
    for (int b = 0; b < BB; ++b) {
        S  += psum[b * TT + t];
        SS += psumsq[b * TT + t];
    }
    const float cnt = (float)(BB * NN * CC);            // 204800
    const float m   = S / cnt;
    const float var = SS / cnt - m * m;
    meanv[t] = m;
    invv[t]  = rsqrtf(var + EPSV);
}

// ---------------------------------------------------------------------------
// Kernel 3: in-place BN apply + ReLU (float4; 4 consecutive c share one t)
// ---------------------------------------------------------------------------
__global__ __launch_bounds__(256) void bn_apply(
    float* __restrict__ y, const float* __restrict__ meanv, const float* __restrict__ invv,
    const float* __restrict__ gamma, const float* __restrict__ beta) {
    const int i4 = blockIdx.x * blockDim.x + threadIdx.x;
    const int total4 = BB * TT * NN * CC / 4;           // 5,120,000
    if (i4 >= total4) return;
    const int t = ((i4 * 4) / (NN * CC)) % TT;
    const float m = meanv[t], iv = invv[t], gm = gamma[t], bt = beta[t];
    float4 v = ((float4*)y)[i4];
    v.x = fmaxf(fmaf((v.x - m) * iv, gm, bt), 0.0f);
    v.y = fmaxf(fmaf((v.y - m) * iv, gm, bt), 0.0f);
    v.z = fmaxf(fmaf((v.z - m) * iv, gm, bt), 0.0f);
    v.w = fmaxf(fmaf((v.w - m) * iv, gm, bt), 0.0f);
    ((float4*)y)[i4] = v;
}

// ---------------------------------------------------------------------------
extern "C" void kernel_launch(void* const* d_in, const int* in_sizes, int n_in,
                              void* d_out, int out_size, void* d_ws, size_t ws_size,
                              hipStream_t stream) {
    const float* x     = (const float*)d_in[0];
    const int*   ei    = (const int*)  d_in[1];
    const float* W1    = (const float*)d_in[2];
    const float* b1    = (const float*)d_in[3];
    const float* W2    = (const float*)d_in[4];
    const float* b2    = (const float*)d_in[5];
    const float* gamma = (const float*)d_in[6];
    const float* beta  = (const float*)d_in[7];
    float* y  = (float*)d_out;
    float* ws = (float*)d_ws;

    float* Ag     = ws;          // 625 floats
    float* psum   = ws + 1024;   // 6400
    float* psumsq = ws + 8192;   // 6400
    float* meanv  = ws + 15360;  // 100
    float* invv   = ws + 15488;  // 100

    build_adj<<<1, 1, 0, stream>>>(ei, Ag);
    gcn_fused<<<NWG, 256, 0, stream>>>(x, W1, b1, W2, b2, Ag, y, psum, psumsq);
    bn_stats<<<1, 128, 0, stream>>>(psum, psumsq, meanv, invv);
    const int total4 = BB * TT * NN * CC / 4;
    bn_apply<<<(total4 + 255) / 256, 256, 0, stream>>>(y, meanv, invv, gamma, beta);
}